// TrueMambaBlock_62362925138582
// MI455X (gfx1250) — compile-verified
//
#include <hip/hip_runtime.h>

// CDNA5 / gfx1250, wave32. FP32 WMMA (V_WMMA_F32_16X16X4_F32) for all dense
// projections; VALU transcendentals for the selective scan; async-to-LDS
// staging (ASYNCcnt) in the final projection kernel.

typedef __attribute__((ext_vector_type(2))) float v2f;
typedef __attribute__((ext_vector_type(8))) float v8f;

#define LSEQ   256   // sequence length (H == W == 256)
#define CMODEL 64    // D_MODEL
#define DIN    128   // D_INNER
#define NSTATE 16    // D_STATE
#define SX     129   // padded LDS row stride for s_x / s_z (conflict-free)
#define SD     37    // padded LDS row stride for s_dbl
#define NPIX   65536 // H*W

__device__ __forceinline__ v8f wmma_f32(v2f a, v2f b, v8f c) {
  // D = A(16x4,f32) * B(4x16,f32) + C(16x16,f32)
  return __builtin_amdgcn_wmma_f32_16x16x4_f32(false, a, false, b, (short)0, c,
                                               false, false);
}

__device__ __forceinline__ float silu_f(float v) {
  return v / (1.f + __expf(-v));
}

// ---------------------------------------------------------------- reductions
__global__ __launch_bounds__(256) void reduce1_kernel(
    const float* __restrict__ x, float* __restrict__ psum,
    float* __restrict__ psq) {
  __shared__ float ss[256], qq[256];
  int tid = threadIdx.x;
  const float4* x4 = (const float4*)x;
  int base4 = blockIdx.x * 1024;
  float s = 0.f, q = 0.f;
  for (int j = 0; j < 4; ++j) {
    float4 v = x4[base4 + j * 256 + tid];
    s += v.x + v.y + v.z + v.w;
    q += v.x * v.x + v.y * v.y + v.z * v.z + v.w * v.w;
  }
  ss[tid] = s; qq[tid] = q;
  __syncthreads();
  for (int off = 128; off > 0; off >>= 1) {
    if (tid < off) { ss[tid] += ss[tid + off]; qq[tid] += qq[tid + off]; }
    __syncthreads();
  }
  if (tid == 0) { psum[blockIdx.x] = ss[0]; psq[blockIdx.x] = qq[0]; }
}

__global__ __launch_bounds__(256) void reduce2_kernel(
    const float* __restrict__ psum, const float* __restrict__ psq,
    float* __restrict__ stats) {
  __shared__ float ss[256], qq[256];
  int tid = threadIdx.x;
  float s = 0.f, q = 0.f;
  for (int j = 0; j < 4; ++j) { s += psum[tid * 4 + j]; q += psq[tid * 4 + j]; }
  ss[tid] = s; qq[tid] = q;
  __syncthreads();
  for (int off = 128; off > 0; off >>= 1) {
    if (tid < off) { ss[tid] += ss[tid + off]; qq[tid] += qq[tid + off]; }
    __syncthreads();
  }
  if (tid == 0) {
    const float invN = 1.f / 4194304.f;
    float mean = ss[0] * invN;
    float var = qq[0] * invN - mean * mean;
    stats[0] = mean;
    stats[1] = rsqrtf(var + 1e-5f);
  }
}

// ------------------------------------------------- normalize + relayout (x2)
__global__ __launch_bounds__(256) void norm_kernel(
    const float* __restrict__ x, const float* __restrict__ gw,
    const float* __restrict__ gb, const float* __restrict__ stats,
    float* __restrict__ yrow, float* __restrict__ ycol) {
  int i = blockIdx.x * 256 + threadIdx.x;  // over 64*65536 elements, (c,h,w)
  int c = i >> 16;
  int hw = i & 65535;
  int h = hw >> 8;
  int w = hw & 255;
  float v = (x[i] - stats[0]) * stats[1] * gw[c] + gb[c];
  yrow[hw * CMODEL + c] = v;                 // (h,w,c): row sequences
  ycol[((w << 8) | h) * CMODEL + c] = v;     // (w,h,c): col sequences
}

// ------------------------------------------------------------- mamba per seq
// One block per (axis, sequence); processes fwd then bwd, merges 0.5*(f+b).
// LDS: s_x 132KB + s_z 132KB + s_dbl 37KB = 302,080 B (<320KB/WG per ISA).
__global__ __launch_bounds__(256) void mamba_kernel(
    const float* __restrict__ yrow, const float* __restrict__ ycol,
    const float* __restrict__ in_w_r, const float* __restrict__ conv_w_r,
    const float* __restrict__ conv_b_r, const float* __restrict__ xproj_w_r,
    const float* __restrict__ dt_w_r, const float* __restrict__ dt_b_r,
    const float* __restrict__ A_log_r, const float* __restrict__ Dp_r,
    const float* __restrict__ out_w_r,
    const float* __restrict__ in_w_c, const float* __restrict__ conv_w_c,
    const float* __restrict__ conv_b_c, const float* __restrict__ xproj_w_c,
    const float* __restrict__ dt_w_c, const float* __restrict__ dt_b_c,
    const float* __restrict__ A_log_c, const float* __restrict__ Dp_c,
    const float* __restrict__ out_w_c,
    float* __restrict__ rowbuf, float* __restrict__ colbuf) {
  __shared__ float s_x[LSEQ * SX];    // x (post conv+silu), then gated y
  __shared__ float s_z[LSEQ * SX];    // z half of the input projection
  __shared__ float s_dbl[LSEQ * SD];  // x_dbl: [0:4]=dt-rank, [4:20]=B, [20:36]=C

  const int tid = threadIdx.x;
  const int wave = tid >> 5;
  const int lane = tid & 31;
  const int lm = lane & 15;   // M / N lane index
  const int lh = lane >> 4;   // K half selector
  const int blk = blockIdx.x;
  const int axis = blk >> 8;  // 0=row, 1=col
  const int sidx = blk & 255;

  const float* seqb   = (axis ? ycol : yrow) + sidx * (LSEQ * CMODEL);
  const float* in_w   = axis ? in_w_c   : in_w_r;
  const float* conv_w = axis ? conv_w_c : conv_w_r;
  const float* conv_b = axis ? conv_b_c : conv_b_r;
  const float* xproj_w= axis ? xproj_w_c: xproj_w_r;
  const float* dt_w   = axis ? dt_w_c   : dt_w_r;
  const float* dt_b   = axis ? dt_b_c   : dt_b_r;
  const float* A_log  = axis ? A_log_c  : A_log_r;
  const float* Dp     = axis ? Dp_c     : Dp_r;
  const float* out_w  = axis ? out_w_c  : out_w_r;

  float* obase;
  int ostride;
  if (axis == 0) { obase = rowbuf + sidx * (LSEQ * CMODEL); ostride = CMODEL; }
  else           { obase = colbuf + sidx * CMODEL;          ostride = LSEQ * CMODEL; }

  for (int dir = 0; dir < 2; ++dir) {
    // ---- GEMM1: xz[256x256] = seq[256x64] @ in_w^T ; x -> s_x, z -> s_z.
    // Each wave owns 2 N-tiles: hoist the B fragment into registers once,
    // reuse across all 16 M-tiles (halves global load traffic).
    {
      const int nt0 = wave << 1;
      for (int ni = 0; ni < 2; ++ni) {
        const int n0 = (nt0 + ni) << 4;
        const float* bp = in_w + (n0 + lm) * CMODEL;
        v2f breg[16];
#pragma unroll
        for (int k = 0; k < 16; ++k) {
          breg[k].x = bp[4 * k + 2 * lh];
          breg[k].y = bp[4 * k + 2 * lh + 1];
        }
        for (int mt = 0; mt < 16; ++mt) {
          const int m0 = mt << 4;
          const int t = m0 + lm;
          const int tt = dir ? (LSEQ - 1 - t) : t;
          const float* ap = seqb + tt * CMODEL;
          v8f acc = {0.f, 0.f, 0.f, 0.f, 0.f, 0.f, 0.f, 0.f};
#pragma unroll
          for (int k = 0; k < 16; ++k) {
            v2f a; a.x = ap[4 * k + 2 * lh]; a.y = ap[4 * k + 2 * lh + 1];
            acc = wmma_f32(a, breg[k], acc);
          }
          if (n0 < DIN) {
#pragma unroll
            for (int r = 0; r < 8; ++r)
              s_x[(m0 + r + 8 * lh) * SX + n0 + lm] = acc[r];
          } else {
#pragma unroll
            for (int r = 0; r < 8; ++r)
              s_z[(m0 + r + 8 * lh) * SX + (n0 - DIN + lm)] = acc[r];
          }
        }
      }
    }
    __syncthreads();

    // ---- depthwise causal conv (K=4) + SiLU, in place (register window)
    if (tid < DIN) {
      const int d = tid;
      const float c0 = conv_w[d * 4 + 0], c1 = conv_w[d * 4 + 1];
      const float c2 = conv_w[d * 4 + 2], c3 = conv_w[d * 4 + 3];
      const float cb = conv_b[d];
      float w3 = 0.f, w2 = 0.f, w1 = 0.f;
      for (int t = 0; t < LSEQ; ++t) {
        const float r0 = s_x[t * SX + d];
        const float v = c0 * w3 + c1 * w2 + c2 * w1 + c3 * r0 + cb;
        s_x[t * SX + d] = silu_f(v);
        w3 = w2; w2 = w1; w1 = r0;
      }
    }
    __syncthreads();

    // ---- GEMM2: x_dbl[256x36] = x[256x128] @ xproj_w^T (N padded to 48)
    for (int tile = wave; tile < 48; tile += 8) {
      const int m0 = (tile & 15) << 4;
      const int n0 = (tile >> 4) << 4;
      const int n = n0 + lm;
      v8f acc = {0.f, 0.f, 0.f, 0.f, 0.f, 0.f, 0.f, 0.f};
      for (int k0 = 0; k0 < DIN; k0 += 4) {
        const int kk = k0 + 2 * lh;
        v2f a; a.x = s_x[(m0 + lm) * SX + kk]; a.y = s_x[(m0 + lm) * SX + kk + 1];
        v2f b;
        if (n < 36) { b.x = xproj_w[n * DIN + kk]; b.y = xproj_w[n * DIN + kk + 1]; }
        else        { b.x = 0.f; b.y = 0.f; }
        acc = wmma_f32(a, b, acc);
      }
      if (n < 36) {
#pragma unroll
        for (int r = 0; r < 8; ++r)
          s_dbl[(m0 + r + 8 * lh) * SD + n] = acc[r];
      }
    }
    __syncthreads();

    // ---- selective scan: thread d owns channel d, 16 states in registers
    if (tid < DIN) {
      const int d = tid;
      float Am[NSTATE];
#pragma unroll
      for (int n = 0; n < NSTATE; ++n) Am[n] = -__expf(A_log[d * NSTATE + n]);
      const float dw0 = dt_w[d * 4 + 0], dw1 = dt_w[d * 4 + 1];
      const float dw2 = dt_w[d * 4 + 2], dw3 = dt_w[d * 4 + 3];
      const float dtb = dt_b[d];
      const float Dd = Dp[d];
      float h[NSTATE];
#pragma unroll
      for (int n = 0; n < NSTATE; ++n) h[n] = 0.f;
      for (int t = 0; t < LSEQ; ++t) {
        const float* db = &s_dbl[t * SD];
        float dtv = db[0] * dw0 + db[1] * dw1 + db[2] * dw2 + db[3] * dw3 + dtb;
        dtv = (dtv > 20.f) ? dtv : __logf(1.f + __expf(dtv));  // softplus
        const float xv = s_x[t * SX + d];
        const float dx = dtv * xv;
        float y = xv * Dd;
#pragma unroll
        for (int n = 0; n < NSTATE; ++n) {
          h[n] = __expf(dtv * Am[n]) * h[n] + dx * db[4 + n];
          y += h[n] * db[20 + n];
        }
        const float zv = s_z[t * SX + d];
        y *= silu_f(zv);
        s_x[t * SX + d] = y;  // overwrite x with gated y
      }
    }
    __syncthreads();

    // ---- GEMM3: out[256x64] = y[256x128] @ out_w^T ; merge 0.5*(fwd+bwd).
    // Each N-tile is owned by a wave pair; B fragment hoisted to registers.
    {
      const int n0 = (wave >> 1) << 4;
      const float* bp = out_w + (n0 + lm) * DIN;
      v2f breg[32];
#pragma unroll
      for (int k = 0; k < 32; ++k) {
        breg[k].x = bp[4 * k + 2 * lh];
        breg[k].y = bp[4 * k + 2 * lh + 1];
      }
      for (int mi = 0; mi < 8; ++mi) {
        const int m0 = (((wave & 1) << 3) + mi) << 4;
        v8f acc = {0.f, 0.f, 0.f, 0.f, 0.f, 0.f, 0.f, 0.f};
#pragma unroll
        for (int k = 0; k < 32; ++k) {
          v2f a;
          a.x = s_x[(m0 + lm) * SX + 4 * k + 2 * lh];
          a.y = s_x[(m0 + lm) * SX + 4 * k + 2 * lh + 1];
          acc = wmma_f32(a, breg[k], acc);
        }
#pragma unroll
        for (int r = 0; r < 8; ++r) {
          const int t = m0 + r + 8 * lh;
          const int tt = dir ? (LSEQ - 1 - t) : t;
          float* p = obase + tt * ostride + n0 + lm;
          if (dir == 0) *p = 0.5f * acc[r];
          else          *p += 0.5f * acc[r];
        }
      }
    }
    __syncthreads();
  }
}

// ------------------------------------- final: residual + proj @ 0.5*(row+col)
// Stages the block's row/col tiles into LDS once via async-to-LDS (each pixel
// row is consumed by 4 N-tiles), XOR-swizzled per 16B granule so the 16-lane
// column reads for the WMMA A operand are bank-conflict-free.
__global__ __launch_bounds__(256) void proj_kernel(
    const float* __restrict__ xin, const float* __restrict__ proj_w,
    const float* __restrict__ rowbuf, const float* __restrict__ colbuf,
    float* __restrict__ out) {
  __shared__ __align__(16) float s_row[128 * CMODEL];
  __shared__ __align__(16) float s_col[128 * CMODEL];
  const int tid = threadIdx.x;
  const int wave = tid >> 5;
  const int lane = tid & 31;
  const int lm = lane & 15;
  const int lh = lane >> 4;
  const int p0 = blockIdx.x * 128;  // 128 pixels per block

  // async copy: 2048 granules (16B) per buffer, 8 per thread
  for (int j = 0; j < 8; ++j) {
    const int g = j * 256 + tid;
    const int pl = g >> 4;            // local pixel 0..127
    const int c4 = g & 15;            // channel granule 0..15
    const int dst = pl * CMODEL + ((c4 ^ (pl & 15)) << 2);
    const size_t src = (size_t)(p0 + pl) * CMODEL + (c4 << 2);
    unsigned ldsr = (unsigned)(uintptr_t)(&s_row[dst]);
    unsigned ldsc = (unsigned)(uintptr_t)(&s_col[dst]);
    const float* gr = rowbuf + src;
    const float* gc = colbuf + src;
    asm volatile("global_load_async_to_lds_b128 %0, %1, off"
                 :: "v"(ldsr), "v"(gr) : "memory");
    asm volatile("global_load_async_to_lds_b128 %0, %1, off"
                 :: "v"(ldsc), "v"(gc) : "memory");
  }
  asm volatile("s_wait_asynccnt 0" ::: "memory");
  __syncthreads();

  for (int tile = wave; tile < 32; tile += 8) {
    const int m0 = (tile & 7) << 4;   // local pixel tile
    const int n0 = (tile >> 3) << 4;
    const int pl = m0 + lm;
    const float* bp = proj_w + (n0 + lm) * CMODEL;
    v8f acc = {0.f, 0.f, 0.f, 0.f, 0.f, 0.f, 0.f, 0.f};
    for (int k0 = 0; k0 < CMODEL; k0 += 4) {
      const int kk = k0 + 2 * lh;
      const int fi = pl * CMODEL + (((kk >> 2) ^ (pl & 15)) << 2) + (kk & 3);
      v2f a;
      a.x = 0.5f * (s_row[fi] + s_col[fi]);
      a.y = 0.5f * (s_row[fi + 1] + s_col[fi + 1]);
      v2f b; b.x = bp[kk]; b.y = bp[kk + 1];
      acc = wmma_f32(a, b, acc);
    }
#pragma unroll
    for (int r = 0; r < 8; ++r) {
      const int pp = p0 + m0 + r + 8 * lh;
      const int o = n0 + lm;
      out[o * NPIX + pp] = xin[o * NPIX + pp] + acc[r];
    }
  }
}

// ---------------------------------------------------------------------------
extern "C" void kernel_launch(void* const* d_in, const int* in_sizes, int n_in,
                              void* d_out, int out_size, void* d_ws,
                              size_t ws_size, hipStream_t stream) {
  (void)in_sizes; (void)n_in; (void)out_size; (void)ws_size;
  // setup_inputs() dict order:
  const float* x      = (const float*)d_in[0];
  const float* gn_w   = (const float*)d_in[1];
  const float* gn_b   = (const float*)d_in[2];
  const float* proj_w = (const float*)d_in[3];
  const float* r_in_w   = (const float*)d_in[4];
  const float* r_conv_w = (const float*)d_in[5];
  const float* r_conv_b = (const float*)d_in[6];
  const float* r_xproj  = (const float*)d_in[7];
  const float* r_dt_w   = (const float*)d_in[8];
  const float* r_dt_b   = (const float*)d_in[9];
  const float* r_A_log  = (const float*)d_in[10];
  const float* r_D      = (const float*)d_in[11];
  const float* r_out_w  = (const float*)d_in[12];
  const float* c_in_w   = (const float*)d_in[13];
  const float* c_conv_w = (const float*)d_in[14];
  const float* c_conv_b = (const float*)d_in[15];
  const float* c_xproj  = (const float*)d_in[16];
  const float* c_dt_w   = (const float*)d_in[17];
  const float* c_dt_b   = (const float*)d_in[18];
  const float* c_A_log  = (const float*)d_in[19];
  const float* c_D      = (const float*)d_in[20];
  const float* c_out_w  = (const float*)d_in[21];
  float* out = (float*)d_out;

  // workspace layout (floats): ~67 MB total
  float* ws = (float*)d_ws;
  float* yrow   = ws;                    // 4,194,304
  float* ycol   = yrow + 4194304;        // 4,194,304
  float* rowbuf = ycol + 4194304;        // 4,194,304
  float* colbuf = rowbuf + 4194304;      // 4,194,304
  float* psum   = colbuf + 4194304;      // 1024
  float* psq    = psum + 1024;           // 1024
  float* stats  = psq + 1024;            // 2 (+pad)

  reduce1_kernel<<<1024, 256, 0, stream>>>(x, psum, psq);
  reduce2_kernel<<<1, 256, 0, stream>>>(psum, psq, stats);
  norm_kernel<<<16384, 256, 0, stream>>>(x, gn_w, gn_b, stats, yrow, ycol);
  mamba_kernel<<<512, 256, 0, stream>>>(
      yrow, ycol,
      r_in_w, r_conv_w, r_conv_b, r_xproj, r_dt_w, r_dt_b, r_A_log, r_D, r_out_w,
      c_in_w, c_conv_w, c_conv_b, c_xproj, c_dt_w, c_dt_b, c_A_log, c_D, c_out_w,
      rowbuf, colbuf);
  proj_kernel<<<512, 256, 0, stream>>>(x, proj_w, rowbuf, colbuf, out);
}